// LSTMDecoder_91087666413992
// MI455X (gfx1250) — compile-verified
//
#include <hip/hip_runtime.h>
#include <hip/hip_bf16.h>

#define H 1024
#define V 50257
#define S 2048

typedef __attribute__((ext_vector_type(2))) float v2f;
typedef __attribute__((ext_vector_type(8))) float v8f;

__device__ __forceinline__ float wave_reduce_sum(float v) {
#pragma unroll
  for (int off = 16; off > 0; off >>= 1)
    v += __shfl_xor(v, off, 32);
  return v;
}

__device__ __forceinline__ float sigmoidf(float x) {
  return 1.0f / (1.0f + expf(-x));
}

// ---------------------------------------------------------------------------
// Kernel 0: hconst = dot(h, attn_W[H:2H]) + attn_b ; zero context accumulator
// ---------------------------------------------------------------------------
__global__ void prep_kernel(const float* __restrict__ h,
                            const float* __restrict__ attn_W,
                            const float* __restrict__ attn_b,
                            float* __restrict__ hconst,
                            float* __restrict__ context) {
  __shared__ float red[256];
  int tid = threadIdx.x;
  float acc = 0.f;
  for (int j = tid; j < H; j += 256) acc += h[j] * attn_W[H + j];
  red[tid] = acc;
  __syncthreads();
  for (int s = 128; s > 0; s >>= 1) {
    if (tid < s) red[tid] += red[tid + s];
    __syncthreads();
  }
  if (tid == 0) hconst[0] = red[0] + attn_b[0];
  for (int j = tid; j < H; j += 256) context[j] = 0.f;
}

// ---------------------------------------------------------------------------
// Kernel 1: coeff[s] = dot(enc[s], attn_W[0:H]) + hconst   (one wave per row)
// ---------------------------------------------------------------------------
__global__ void coeff_kernel(const float* __restrict__ enc,
                             const float* __restrict__ attn_W,
                             const float* __restrict__ hconst,
                             float* __restrict__ coeff) {
  int lane = threadIdx.x & 31;
  int wid  = threadIdx.x >> 5;
  int s = blockIdx.x * 8 + wid;
  const float4* er = (const float4*)(enc + (size_t)s * H);
  const float4* wr = (const float4*)attn_W;
  float acc = 0.f;
#pragma unroll
  for (int it = 0; it < 8; ++it) {
    int idx = it * 32 + lane;           // H/4 = 256 float4 per row
    float4 e = er[idx];
    float4 w = wr[idx];
    acc += e.x * w.x + e.y * w.y + e.z * w.z + e.w * w.w;
  }
  acc = wave_reduce_sum(acc);
  if (lane == 0) coeff[s] = acc + hconst[0];
}

// ---------------------------------------------------------------------------
// Kernel 2: context[j] += sum_{s in chunk} enc[s][j] * coeff[s]
// grid = (H/256, 16); coalesced column access, atomic f32 accumulate
// ---------------------------------------------------------------------------
__global__ void context_kernel(const float* __restrict__ enc,
                               const float* __restrict__ coeff,
                               float* __restrict__ context) {
  int j  = blockIdx.x * 256 + threadIdx.x;
  int s0 = blockIdx.y * (S / 16);
  float acc = 0.f;
  for (int s = s0; s < s0 + S / 16; ++s)
    acc += enc[(size_t)s * H + j] * coeff[s];
  atomicAdd(&context[j], acc);
}

// ---------------------------------------------------------------------------
// Kernel 3: z[i] = relu( dot([x|context], comb_W[i]) + comb_b[i] )
// one wave per output element
// ---------------------------------------------------------------------------
__global__ void combined_kernel(const float* __restrict__ embedding,
                                const int* __restrict__ ids,
                                const float* __restrict__ context,
                                const float* __restrict__ comb_W,
                                const float* __restrict__ comb_b,
                                float* __restrict__ z) {
  int lane = threadIdx.x & 31;
  int wid  = threadIdx.x >> 5;
  int i = blockIdx.x * 8 + wid;
  const float4* w  = (const float4*)(comb_W + (size_t)i * (2 * H));
  const float4* x4 = (const float4*)(embedding + (size_t)ids[0] * H);
  const float4* c4 = (const float4*)context;
  float acc = 0.f;
#pragma unroll
  for (int it = 0; it < 16; ++it) {
    int idx = it * 32 + lane;           // 0..511 float4 over 2H
    float4 wv = w[idx];
    float4 vv = (idx < 256) ? x4[idx] : c4[idx - 256];   // uniform per it
    acc += wv.x * vv.x + wv.y * vv.y + wv.z * vv.z + wv.w * vv.w;
  }
  acc = wave_reduce_sum(acc);
  if (lane == 0) z[i] = fmaxf(acc + comb_b[i], 0.f);
}

// ---------------------------------------------------------------------------
// Kernel 4: gates = W_ih@z + b_ih + W_hh@h + b_hh  via v_wmma_f32_16x16x4_f32
// One wave computes 16 rows. A = 16x4 weight tile (lanes m<16 hold K={0,1},
// lanes 16.. hold K={2,3} per ISA 32-bit A layout). B column N=0 carries the
// activation vector: lane0 VGPR{0,1}=K{0,1}, lane16 VGPR{0,1}=K{2,3}.
// B is built BRANCHLESSLY: every lane does the (broadcast) 8B load and lanes
// with m!=0 are zeroed via a hoisted multiplicative mask -> no exec juggling,
// EXEC stays all-ones through the WMMA loop.
// Result column 0: lane0 VGPR0..7 = rows 0..7, lane16 VGPR0..7 = rows 8..15.
// ---------------------------------------------------------------------------
__global__ void gates_wmma_kernel(const float* __restrict__ W_ih,
                                  const float* __restrict__ W_hh,
                                  const float* __restrict__ b_ih,
                                  const float* __restrict__ b_hh,
                                  const float* __restrict__ z,
                                  const float* __restrict__ h,
                                  float* __restrict__ gates) {
  int lane = threadIdx.x & 31;
  int wid  = threadIdx.x >> 5;
  int row0 = (blockIdx.x * 8 + wid) * 16;
  int half = lane >> 4;                 // 0: K=0,1   1: K=2,3
  int m    = lane & 15;                 // A-matrix row within tile
  float msk = (m == 0) ? 1.0f : 0.0f;   // column-0 lane mask (hoisted)

  const float* Arow = W_ih + (size_t)(row0 + m) * H + half * 2;
  const float* Brow = W_hh + (size_t)(row0 + m) * H + half * 2;
  const float* zv = z + half * 2;
  const float* hv = h + half * 2;

  v8f c = {};

  // Phase 1: W_ih x z
#pragma unroll 8
  for (int kk = 0; kk < H; kk += 4) {
    v2f a = *(const v2f*)(Arow + kk);
    v2f bv = *(const v2f*)(zv + kk);    // broadcast load, all lanes active
    v2f b;
    b.x = bv.x * msk;
    b.y = bv.y * msk;
    c = __builtin_amdgcn_wmma_f32_16x16x4_f32(false, a, false, b, (short)0, c,
                                              false, false);
  }
  // Phase 2: W_hh x h (same accumulator)
#pragma unroll 8
  for (int kk = 0; kk < H; kk += 4) {
    v2f a = *(const v2f*)(Brow + kk);
    v2f bv = *(const v2f*)(hv + kk);
    v2f b;
    b.x = bv.x * msk;
    b.y = bv.y * msk;
    c = __builtin_amdgcn_wmma_f32_16x16x4_f32(false, a, false, b, (short)0, c,
                                              false, false);
  }

  if (m == 0) {
    int rbase = row0 + half * 8;
#pragma unroll
    for (int r = 0; r < 8; ++r) {
      int g = rbase + r;
      gates[g] = c[r] + b_ih[g] + b_hh[g];
    }
  }
}

// ---------------------------------------------------------------------------
// Kernel 5: LSTM cell elementwise -> h_new, c_new (written straight to d_out)
// ---------------------------------------------------------------------------
__global__ void cell_kernel(const float* __restrict__ gates,
                            const float* __restrict__ c0,
                            float* __restrict__ h_new,
                            float* __restrict__ c_new) {
  int j = blockIdx.x * 256 + threadIdx.x;
  float ig = sigmoidf(gates[j]);
  float fg = sigmoidf(gates[H + j]);
  float gg = tanhf(gates[2 * H + j]);
  float og = sigmoidf(gates[3 * H + j]);
  float cn = fg * c0[j] + ig * gg;
  c_new[j] = cn;
  h_new[j] = og * tanhf(cn);
}

// ---------------------------------------------------------------------------
// Kernel 6: logits[v] = dot(h_new, out_W[v]) + out_b[v]
// Dominant 206MB stream: one wave per row, float4 fully-coalesced loads.
// ---------------------------------------------------------------------------
__global__ void logits_kernel(const float* __restrict__ out_W,
                              const float* __restrict__ out_b,
                              const float* __restrict__ h_new,
                              float* __restrict__ logits) {
  int lane = threadIdx.x & 31;
  int wid  = threadIdx.x >> 5;
  int v = blockIdx.x * 8 + wid;
  if (v >= V) return;
  const float4* w  = (const float4*)(out_W + (size_t)v * H);
  const float4* hh = (const float4*)h_new;
  float acc = 0.f;
#pragma unroll
  for (int it = 0; it < 8; ++it) {
    int idx = it * 32 + lane;
    float4 wv = w[idx];
    float4 hv = hh[idx];
    acc += wv.x * hv.x + wv.y * hv.y + wv.z * hv.z + wv.w * hv.w;
  }
  acc = wave_reduce_sum(acc);
  if (lane == 0) logits[v] = acc + out_b[v];
}

// ---------------------------------------------------------------------------
// Kernel 7: in-place log_softmax over V (single 1024-thread workgroup)
// ---------------------------------------------------------------------------
__global__ void logsoftmax_kernel(float* __restrict__ logits) {
  __shared__ float red[1024];
  __shared__ float stat[2];
  int tid = threadIdx.x;

  float m = -INFINITY;
  for (int v = tid; v < V; v += 1024) m = fmaxf(m, logits[v]);
  red[tid] = m;
  __syncthreads();
  for (int s = 512; s > 0; s >>= 1) {
    if (tid < s) red[tid] = fmaxf(red[tid], red[tid + s]);
    __syncthreads();
  }
  if (tid == 0) stat[0] = red[0];
  __syncthreads();
  m = stat[0];

  float sum = 0.f;
  for (int v = tid; v < V; v += 1024) sum += expf(logits[v] - m);
  red[tid] = sum;
  __syncthreads();
  for (int s = 512; s > 0; s >>= 1) {
    if (tid < s) red[tid] += red[tid + s];
    __syncthreads();
  }
  if (tid == 0) stat[1] = logf(red[0]);
  __syncthreads();

  float shift = m + stat[1];
  for (int v = tid; v < V; v += 1024) logits[v] -= shift;
}

// ---------------------------------------------------------------------------
extern "C" void kernel_launch(void* const* d_in, const int* in_sizes, int n_in,
                              void* d_out, int out_size, void* d_ws,
                              size_t ws_size, hipStream_t stream) {
  const int*   input_ids = (const int*)d_in[0];
  const float* h0        = (const float*)d_in[1];
  const float* c0        = (const float*)d_in[2];
  const float* enc       = (const float*)d_in[3];
  const float* embedding = (const float*)d_in[4];
  const float* attn_W    = (const float*)d_in[5];
  const float* attn_b    = (const float*)d_in[6];
  const float* comb_W    = (const float*)d_in[7];
  const float* comb_b    = (const float*)d_in[8];
  const float* W_ih      = (const float*)d_in[9];
  const float* W_hh      = (const float*)d_in[10];
  const float* b_ih      = (const float*)d_in[11];
  const float* b_hh      = (const float*)d_in[12];
  const float* out_W     = (const float*)d_in[13];
  const float* out_b     = (const float*)d_in[14];
  (void)in_sizes; (void)n_in; (void)out_size; (void)ws_size;

  float* out = (float*)d_out;
  float* ws  = (float*)d_ws;

  // workspace layout (floats): hconst[16] | coeff[S] | context[H] | z[H] | gates[4H]
  float* hconst  = ws;
  float* coeff   = ws + 16;
  float* context = coeff + S;
  float* zbuf    = context + H;
  float* gates   = zbuf + H;

  // output layout: logp[V] | h_new[H] | c_new[H]
  float* logits = out;
  float* h_new  = out + V;
  float* c_new  = out + V + H;

  prep_kernel<<<1, 256, 0, stream>>>(h0, attn_W, attn_b, hconst, context);
  coeff_kernel<<<S / 8, 256, 0, stream>>>(enc, attn_W, hconst, coeff);
  context_kernel<<<dim3(H / 256, 16), 256, 0, stream>>>(enc, coeff, context);
  combined_kernel<<<H / 8, 256, 0, stream>>>(embedding, input_ids, context,
                                             comb_W, comb_b, zbuf);
  gates_wmma_kernel<<<(4 * H / 16) / 8, 256, 0, stream>>>(W_ih, W_hh, b_ih,
                                                          b_hh, zbuf, h0, gates);
  cell_kernel<<<H / 256, 256, 0, stream>>>(gates, c0, h_new, c_new);
  logits_kernel<<<(V + 7) / 8, 256, 0, stream>>>(out_W, out_b, h_new, logits);
  logsoftmax_kernel<<<1, 1024, 0, stream>>>(logits);
}